// PositionalEncodingLayer_34574486732876
// MI455X (gfx1250) — compile-verified
//
#include <hip/hip_runtime.h>
#include <cmath>

// out[b,t,d] = X[b,t,d] + PE[t,d]
// PE[t, 2i]   = sin(t * 10000^(-2i/D))
// PE[t, 2i+1] = cos(t * 10000^(-2i/D))
//
// X: (B=8, T=4096, D=1024) fp32. Pure streaming op: 268 MB traffic,
// ~11.5 us floor at 23.3 TB/s. PE depends only on (t,d) -> compute once
// per thread (2 precise sincos), reuse across all 8 batches. Non-temporal
// b128 loads/stores since the stream exceeds the 192 MB L2 and nothing
// is re-read. Dims are compile-time so no fp divide / runtime muls.

typedef float v4f __attribute__((ext_vector_type(4)));

constexpr int kB = 8;
constexpr int kT = 4096;
constexpr int kD = 1024;

__global__ __launch_bounds__(256) void pe_add_kernel(
    const float* __restrict__ X,
    float* __restrict__ out)
{
    const int t  = blockIdx.x;          // one (T, D) row per block
    const int d0 = threadIdx.x << 2;    // 4 consecutive floats per thread

    // Pair indices covered by this thread's float4: i0 = d0/2, i1 = i0+1.
    // freq_i = 10000^(-2i/D) = exp2( -(2i/D) * log2(10000) )
    constexpr float L2_1E4 = 13.287712379549449f;   // log2(10000)
    constexpr float invD   = 1.0f / (float)kD;      // compile-time constant
    const float i0 = (float)(d0 >> 1);
    const float i1 = i0 + 1.0f;

    const float f0 = exp2f(-(2.0f * invD * L2_1E4) * i0);
    const float f1 = exp2f(-(2.0f * invD * L2_1E4) * i1);

    const float pos = (float)t;
    float s0, c0, s1, c1;
    // Precise sincos: args reach ~4095 rad, so proper range reduction
    // matters for matching the fp32 reference.
    sincosf(pos * f0, &s0, &c0);
    sincosf(pos * f1, &s1, &c1);

    const v4f pe = { s0, c0, s1, c1 };   // interleaved sin/cos layout

    const size_t row     = (size_t)t * (size_t)kD + (size_t)d0;
    constexpr size_t bstride = (size_t)kT * (size_t)kD;

    // Fully unrolled batch loop: 8 independent b128 NT loads in flight,
    // then add + NT store. Amortizes the transcendental work 8x.
    #pragma unroll
    for (int b = 0; b < kB; ++b) {
        const size_t off = row + (size_t)b * bstride;
        const v4f* __restrict__ src = (const v4f*)(X + off);
        v4f*       __restrict__ dst = (v4f*)(out + off);

        v4f x = __builtin_nontemporal_load(src);   // global_load_b128 th:NT
        x += pe;                                   // 4x v_add_f32
        __builtin_nontemporal_store(x, dst);       // global_store_b128 th:NT
    }
}

extern "C" void kernel_launch(void* const* d_in, const int* in_sizes, int n_in,
                              void* d_out, int out_size, void* d_ws, size_t ws_size,
                              hipStream_t stream) {
    (void)n_in; (void)d_ws; (void)ws_size; (void)out_size; (void)in_sizes;

    const float* X  = (const float*)d_in[0];
    float*      out = (float*)d_out;

    dim3 grid(kT);            // one block per t-row
    dim3 block(kD / 4);       // 256 threads, each owning a float4 slice

    pe_add_kernel<<<grid, block, 0, stream>>>(X, out);
}